// VoxelSetAbstraction_78537771975383
// MI455X (gfx1250) — compile-verified
//
#include <hip/hip_runtime.h>
#include <math.h>

typedef float v2f __attribute__((ext_vector_type(2)));
typedef float v8f __attribute__((ext_vector_type(8)));

#define QPB 2  // queries per block in the MLP kernel (1 wave32 per query)

// ---------------------------------------------------------------------------
// Kernel 1: ball query. One wave32 per (query, scale). Scans the query's own
// batch range in index order; wave ballot enumerates in-ball points so the
// first `nsample` smallest indices are kept (== reference top_k(-index)).
// ---------------------------------------------------------------------------
__global__ __launch_bounds__(32) void ball_query_kernel(
    const float* __restrict__ xyz, const float* __restrict__ new_xyz,
    const int* __restrict__ xyz_cnt, const int* __restrict__ new_cnt,
    int nb, int* __restrict__ idx0, int* __restrict__ idx1)
{
    const int m     = blockIdx.x;
    const int scale = blockIdx.y;
    const int ns    = scale ? 32 : 16;
    const float r2  = scale ? (1.6f * 1.6f) : (0.8f * 0.8f);
    int* idx = scale ? (idx1 + (size_t)m * 32) : (idx0 + (size_t)m * 16);

    // batch id of query m, then the point range of that batch
    int b = 0, acc = 0;
    for (int i = 0; i < nb; ++i) { int c = new_cnt[i]; if (m < acc + c) { b = i; break; } acc += c; }
    int ps = 0;
    for (int i = 0; i < b; ++i) ps += xyz_cnt[i];
    const int pe = ps + xyz_cnt[b];

    const float qx = new_xyz[m * 3 + 0];
    const float qy = new_xyz[m * 3 + 1];
    const float qz = new_xyz[m * 3 + 2];

    const int lane = threadIdx.x;
    int count = 0, first = -1;
    for (int base = ps; base < pe && count < ns; base += 32) {
        const int p = base + lane;
        bool ok = false;
        if (p < pe) {
            const float dx = xyz[p * 3 + 0] - qx;
            const float dy = xyz[p * 3 + 1] - qy;
            const float dz = xyz[p * 3 + 2] - qz;
            ok = (dx * dx + dy * dy + dz * dz) < r2;
        }
        const unsigned mask = __builtin_amdgcn_ballot_w32(ok);
        if (ok) {
            const int rank = __popc(mask & ((1u << lane) - 1u));
            if (count + rank < ns) idx[count + rank] = p;
        }
        if (first < 0 && mask != 0u) first = base + (__ffs(mask) - 1);
        count += __popc(mask);
    }
    if (count >= ns) return;
    if (count == 0) {                       // empty ball -> zero rows downstream
        for (int t = lane; t < ns; t += 32) idx[t] = -1;
    } else {                                // pad with first found index
        for (int t = count + lane; t < ns; t += 32) idx[t] = first;
    }
}

// ---------------------------------------------------------------------------
// Kernel 2: gather + 2-layer shared MLP (f32 WMMA 16x16x4) + neighbor maxpool.
// One wave32 per query; ns=16 -> one 16-row A tile, ns=32 -> two.
// A layout (ISA 7.12.2, 32-bit A 16x4): lane&15 = row M; half h: K pair
//   (k0+2h, k0+1+2h). C/D layout: VGPR i, half h -> row i+8h, col lane&15.
// ---------------------------------------------------------------------------
__global__ __launch_bounds__(64) void group_mlp_kernel(
    const float* __restrict__ xyz, const float* __restrict__ feats,
    const float* __restrict__ new_xyz,
    const float* __restrict__ W0, const float* __restrict__ Gm0, const float* __restrict__ Bt0,
    const float* __restrict__ W1, const float* __restrict__ Gm1, const float* __restrict__ Bt1,
    const int* __restrict__ idxbuf, int ns, int scale,
    float* __restrict__ out, int Mtot)
{
    __shared__ float wt0[68 * 64];          // W0^T, K-major, K padded 67->68
    __shared__ float wt1[64 * 64];          // W1^T, K-major
    __shared__ float Gbuf[QPB][32 * 68];    // grouped tile, reused for H after L1

    const int tid = threadIdx.x;
    for (int i = tid; i < 68 * 64; i += 64) {
        const int k = i >> 6, n = i & 63;
        wt0[i] = (k < 67) ? W0[n * 67 + k] : 0.0f;
    }
    for (int i = tid; i < 64 * 64; i += 64) {
        const int k = i >> 6, n = i & 63;
        wt1[i] = W1[n * 64 + k];
    }
    __syncthreads();

    const int w    = tid >> 5;
    const int lane = tid & 31;
    int m = blockIdx.x * QPB + w;
    if (m >= Mtot) m = Mtot - 1;            // duplicate tail work, keep waves uniform

    // -------- gather: rel-xyz (3) + features (64) + zero pad (1) ----------
    const int* idx = idxbuf + (size_t)m * ns;
    const float qx = new_xyz[m * 3 + 0];
    const float qy = new_xyz[m * 3 + 1];
    const float qz = new_xyz[m * 3 + 2];
    for (int r = lane; r < ns; r += 32) {
        const int p = idx[r];
        float* dst = &Gbuf[w][r * 68];
        if (p < 0) {
            #pragma unroll
            for (int c = 0; c < 68; ++c) dst[c] = 0.0f;
        } else {
            dst[0] = xyz[p * 3 + 0] - qx;
            dst[1] = xyz[p * 3 + 1] - qy;
            dst[2] = xyz[p * 3 + 2] - qz;
            const float4* f4 = (const float4*)(feats + (size_t)p * 64);
            #pragma unroll
            for (int j = 0; j < 16; ++j) {
                const float4 v = f4[j];
                dst[3 + 4 * j + 0] = v.x; dst[3 + 4 * j + 1] = v.y;
                dst[3 + 4 * j + 2] = v.z; dst[3 + 4 * j + 3] = v.w;
            }
            dst[67] = 0.0f;
        }
    }
    __syncthreads();

    const int h  = lane >> 4;               // wave half selects K pair
    const int ln = lane & 15;
    const int mtiles = ns >> 4;             // 1 or 2
    const float inv_std = 1.0f / sqrtf(1.0f + 1e-3f);

    // -------- layer 1: D(16x16) = G(16x68) * W0^T(68x16) per N-tile --------
    v8f acc[2][4];
    for (int mt = 0; mt < mtiles; ++mt) {
        const float* Arow = &Gbuf[w][(mt * 16 + ln) * 68];
        for (int t = 0; t < 4; ++t) {
            v8f c = {};
            #pragma unroll
            for (int k0 = 0; k0 < 68; k0 += 4) {
                v2f a, bfr;
                a.x = Arow[k0 + 2 * h];
                a.y = Arow[k0 + 1 + 2 * h];
                const float* bp = &wt0[(k0 + 2 * h) * 64 + t * 16 + ln];
                bfr.x = bp[0];
                bfr.y = bp[64];
                c = __builtin_amdgcn_wmma_f32_16x16x4_f32(
                        false, a, false, bfr, (short)0, c, false, false);
            }
            acc[mt][t] = c;
        }
    }
    __syncthreads();

    // -------- BN(eval)+ReLU, restage H into Gbuf (row i+8h, col lane&15) ---
    for (int mt = 0; mt < mtiles; ++mt) {
        for (int t = 0; t < 4; ++t) {
            const int n = t * 16 + ln;
            const float sc = Gm0[n] * inv_std, bi = Bt0[n];
            #pragma unroll
            for (int i = 0; i < 8; ++i)
                Gbuf[w][(mt * 16 + i + 8 * h) * 68 + n] =
                    fmaxf(acc[mt][t][i] * sc + bi, 0.0f);
        }
    }
    __syncthreads();

    // -------- layer 2 + fused neighbor max-pool ---------------------------
    for (int t = 0; t < 4; ++t) {
        const int n = t * 16 + ln;
        const float sc = Gm1[n] * inv_std, bi = Bt1[n];
        float vmax = -3.402823466e38f;
        for (int mt = 0; mt < mtiles; ++mt) {
            const float* Arow = &Gbuf[w][(mt * 16 + ln) * 68];
            v8f c = {};
            #pragma unroll
            for (int k0 = 0; k0 < 64; k0 += 4) {
                v2f a, bfr;
                a.x = Arow[k0 + 2 * h];
                a.y = Arow[k0 + 1 + 2 * h];
                const float* bp = &wt1[(k0 + 2 * h) * 64 + n];
                bfr.x = bp[0];
                bfr.y = bp[64];
                c = __builtin_amdgcn_wmma_f32_16x16x4_f32(
                        false, a, false, bfr, (short)0, c, false, false);
            }
            #pragma unroll
            for (int i = 0; i < 8; ++i)
                vmax = fmaxf(vmax, fmaxf(c[i] * sc + bi, 0.0f));
        }
        vmax = fmaxf(vmax, __shfl_xor(vmax, 16, 32));  // merge row halves 0-7 / 8-15
        if (lane < 16) out[(size_t)m * 128 + scale * 64 + n] = vmax;
    }
}

// ---------------------------------------------------------------------------
extern "C" void kernel_launch(void* const* d_in, const int* in_sizes, int n_in,
                              void* d_out, int out_size, void* d_ws, size_t ws_size,
                              hipStream_t stream)
{
    (void)n_in; (void)out_size; (void)ws_size;
    const float* xyz     = (const float*)d_in[0];
    const float* feats   = (const float*)d_in[1];
    const float* new_xyz = (const float*)d_in[2];
    const float* w00 = (const float*)d_in[3];
    const float* g00 = (const float*)d_in[4];
    const float* b00 = (const float*)d_in[5];
    const float* w01 = (const float*)d_in[6];
    const float* g01 = (const float*)d_in[7];
    const float* b01 = (const float*)d_in[8];
    const float* w10 = (const float*)d_in[9];
    const float* g10 = (const float*)d_in[10];
    const float* b10 = (const float*)d_in[11];
    const float* w11 = (const float*)d_in[12];
    const float* g11 = (const float*)d_in[13];
    const float* b11 = (const float*)d_in[14];
    const int* xyz_cnt = (const int*)d_in[15];
    const int* new_cnt = (const int*)d_in[16];

    const int Mtot = in_sizes[2] / 3;
    const int nb   = in_sizes[15];

    int* idx0 = (int*)d_ws;                       // Mtot*16 ints
    int* idx1 = idx0 + (size_t)Mtot * 16;         // Mtot*32 ints
    float* out = (float*)d_out;

    dim3 gq(Mtot, 2);
    ball_query_kernel<<<gq, 32, 0, stream>>>(xyz, new_xyz, xyz_cnt, new_cnt,
                                             nb, idx0, idx1);

    dim3 gm((Mtot + QPB - 1) / QPB);
    group_mlp_kernel<<<gm, 64, 0, stream>>>(xyz, feats, new_xyz,
                                            w00, g00, b00, w01, g01, b01,
                                            idx0, 16, 0, out, Mtot);
    group_mlp_kernel<<<gm, 64, 0, stream>>>(xyz, feats, new_xyz,
                                            w10, g10, b10, w11, g11, b11,
                                            idx1, 32, 1, out, Mtot);
}